// bLSTM_13838384627924
// MI455X (gfx1250) — compile-verified
//
#include <hip/hip_runtime.h>

// Problem sizes (fixed by the reference)
#define Bq   256
#define Lq   512
#define Kq   128
#define Hq   512
#define NGq  2048          // 4*H
#define NBLK 64            // persistent grid: 8 batch-blocks x 8 h-slices
#define TPB  256           // 8 wave32s
#define KCH  32            // K-chunk staged per TDM round
#define NCH_IH (Kq / KCH)              // 4
#define NCH_HH (Hq / KCH)              // 16
#define NCH    (NCH_IH + NCH_HH)       // 20
#define WROW 72            // LDS row stride (dwords) for weight tiles (64 + 8 pad)
#define AROW 36            // LDS row stride (dwords) for A tiles (32 + 4 pad)

typedef __attribute__((ext_vector_type(2))) float v2f;
typedef __attribute__((ext_vector_type(8))) float v8f;
typedef __attribute__((ext_vector_type(4))) unsigned int v4u;
typedef __attribute__((ext_vector_type(8))) int v8i;
typedef __attribute__((ext_vector_type(4))) int v4i;

// ---------------------------------------------------------------------------
// Workspace init: zero h double-buffer, barrier counters, loss slot.
// ---------------------------------------------------------------------------
__global__ void init_ws_kernel(float* hbuf, unsigned* bar, float* loss_slot) {
    int i = blockIdx.x * blockDim.x + threadIdx.x;
    int n = 2 * Bq * Hq;
    for (int j = i; j < n; j += gridDim.x * blockDim.x) hbuf[j] = 0.0f;
    if (i == 0) { bar[0] = 0u; bar[1] = 0u; *loss_slot = 0.0f; }
}

// ---------------------------------------------------------------------------
// Bayes-by-backprop loss: elementwise over (w_hh, w_ih, bias) params.
// ---------------------------------------------------------------------------
__device__ __forceinline__ float bayes_elem(float p, float mu, float rho) {
    const float NLOG_SQRT_2PI = -0.91893853320467274f;
    float lp1 = NLOG_SQRT_2PI - 0.5f * p * p;
    float lp2 = NLOG_SQRT_2PI + 6.90775527898213705f - p * p * 500000.0f;
    float prior = logf(0.5f * __expf(lp1) + 0.5f * __expf(lp2));
    float sigma = log1pf(__expf(rho));                 // softplus(rho)
    float d = p - mu;
    float vari = -0.57236494292470008f - logf(sigma) - d * d / (2.0f * sigma * sigma);
    return vari - prior;
}

__global__ void bayes_loss_kernel(const float* __restrict__ whh,
                                  const float* __restrict__ whh_mu,
                                  const float* __restrict__ whh_rho,
                                  const float* __restrict__ wih,
                                  const float* __restrict__ wih_mu,
                                  const float* __restrict__ wih_rho,
                                  const float* __restrict__ bb,
                                  const float* __restrict__ b_mu,
                                  const float* __restrict__ b_rho,
                                  float* loss) {
    const int N1 = Hq * NGq, N2 = Kq * NGq, N3 = NGq;
    const int total = N1 + N2 + N3;
    float acc = 0.0f;
    for (int i = blockIdx.x * blockDim.x + threadIdx.x; i < total;
         i += gridDim.x * blockDim.x) {
        float v;
        if (i < N1) {
            v = bayes_elem(whh[i], whh_mu[i], whh_rho[i]);
        } else if (i < N1 + N2) {
            int j = i - N1;
            v = bayes_elem(wih[j], wih_mu[j], wih_rho[j]);
        } else {
            int j = i - N1 - N2;
            v = bayes_elem(bb[j], b_mu[j], b_rho[j]);
        }
        acc += v;
    }
    __shared__ float red[TPB];
    red[threadIdx.x] = acc;
    __syncthreads();
    for (int s = TPB / 2; s > 0; s >>= 1) {
        if (threadIdx.x < s) red[threadIdx.x] += red[threadIdx.x + s];
        __syncthreads();
    }
    if (threadIdx.x == 0) atomicAdd(loss, red[0]);
}

// ---------------------------------------------------------------------------
// Grid-wide barrier (all NBLK blocks co-resident; persistent kernel).
// ---------------------------------------------------------------------------
__device__ __forceinline__ void grid_barrier(unsigned* cnt, unsigned* gen) {
    __threadfence();
    __syncthreads();
    if (threadIdx.x == 0) {
        unsigned g = __hip_atomic_load(gen, __ATOMIC_RELAXED, __HIP_MEMORY_SCOPE_AGENT);
        unsigned arrived =
            __hip_atomic_fetch_add(cnt, 1u, __ATOMIC_ACQ_REL, __HIP_MEMORY_SCOPE_AGENT);
        if (arrived == (unsigned)(NBLK - 1)) {
            __hip_atomic_store(cnt, 0u, __ATOMIC_RELAXED, __HIP_MEMORY_SCOPE_AGENT);
            __hip_atomic_fetch_add(gen, 1u, __ATOMIC_RELEASE, __HIP_MEMORY_SCOPE_AGENT);
        } else {
            while (__hip_atomic_load(gen, __ATOMIC_ACQUIRE, __HIP_MEMORY_SCOPE_AGENT) == g) {
                __builtin_amdgcn_s_sleep(2);
            }
        }
    }
    __syncthreads();
}

// ---------------------------------------------------------------------------
// Tensor Data Mover: DMA a 2D tile (tile_rows x tile_cols f32, row stride in
// elements) from global memory into LDS at byte offset lds_off, padding each
// row from tile_cols dwords to (tile_cols + pad) dwords in LDS.
//   pad_interval_code i: pad after every 2^(i+1) dwords
//   pad_amount_code  a: pad by (a+1) dwords
// NOTE: this toolchain exposes the 6-arg builtin
//   (uint32x4 g0, int32x8 g1, int32x4, int32x4, int32x8, i32 cpol)
// ---------------------------------------------------------------------------
__device__ __forceinline__ void tdm_load_2d(unsigned lds_off, const float* gsrc,
                                            int tile_cols, int tile_rows,
                                            long long row_stride_elems,
                                            int pad_interval_code,
                                            int pad_amount_code) {
    unsigned long long ga = (unsigned long long)(uintptr_t)gsrc;
    unsigned long long st = (unsigned long long)row_stride_elems;
    v4u g0;
    g0[0] = 1u;                                        // count=1, user descriptor
    g0[1] = lds_off;                                   // LDS byte address
    g0[2] = (unsigned)(ga & 0xFFFFFFFFull);            // global_addr[31:0]
    g0[3] = (unsigned)((ga >> 32) & 0x01FFFFFFull)     // global_addr[56:32]
            | 0x80000000u;                             // type=2 ("image")
    v8i g1;
    g1[0] = (2 << 16)                                  // data_size = 4B
          | (1 << 20)                                  // pad_enable
          | (pad_interval_code << 22)
          | (pad_amount_code << 25);
    g1[1] = 0;                                         // tensor_dim0[15:0] = 0 (of 2^30)
    g1[2] = 0x4000;                                    // tensor_dim0 hi | tensor_dim1 lo
    g1[3] = 0x4000 | (tile_cols << 16);                // tensor_dim1 hi | tile_dim0
    g1[4] = tile_rows;                                 // tile_dim1 | tile_dim2=0
    g1[5] = (int)(st & 0xFFFFFFFFull);                 // tensor_dim0_stride[31:0]
    g1[6] = (int)((st >> 32) & 0xFFFFull);             // stride[47:32] | dim1_stride lo=0
    g1[7] = 0;
    v4i z4;
    z4[0] = 0; z4[1] = 0; z4[2] = 0; z4[3] = 0;
    v8i z8;
    z8[0] = 0; z8[1] = 0; z8[2] = 0; z8[3] = 0;
    z8[4] = 0; z8[5] = 0; z8[6] = 0; z8[7] = 0;
    __builtin_amdgcn_tensor_load_to_lds(g0, g1, z4, z4, z8, 0);
}

// ---------------------------------------------------------------------------
// Persistent fused Bayesian-LSTM recurrence.
//   WG (64 total) = (32-row batch block) x (64-col h slice, all 4 gate groups)
//   wave (8/WG)   = 16x64 gate tile: m-tile = wid&1, gate group = wid>>1
//   per step: 20 K-chunks of 32; wave 0 TDM-stages 4 weight tiles + 1 A tile
//   per chunk into double-buffered LDS (overlapped with WMMA on the previous
//   chunk); gates -> LDS; cell update with c in registers; h double-buffered
//   in workspace; one grid barrier per timestep.
// ---------------------------------------------------------------------------
__global__ void __launch_bounds__(TPB, 1)
blstm_kernel(const float* __restrict__ x, const float* __restrict__ whh,
             const float* __restrict__ wih, const float* __restrict__ bias,
             float* __restrict__ out_hidden, float* __restrict__ out_hT,
             float* __restrict__ out_cT, float* __restrict__ hbuf0,
             float* __restrict__ hbuf1, unsigned* bar) {
    __shared__ float gate_s[32][260];            // 32 rows x 256 gate cols (+pad)
    __shared__ float wbuf[2][4][KCH * WROW];     // weight tiles: [buf][grp][32x72]
    __shared__ float abuf[2][KCH * AROW];        // A tiles:      [buf][32x36]

    const int lane = threadIdx.x & 31;
    const int wid  = threadIdx.x >> 5;
    const int lm   = lane & 15;             // A: M row within tile
    const int ln   = lane & 15;             // B: N col within tile
    const int kh   = (lane >> 4) << 1;      // K sub-offset (0 or 2)

    const int hs    = (blockIdx.x & 7) * 64;       // h-slice base (64 cols)
    const int m0wg  = (blockIdx.x >> 3) * 32;      // batch block base (32 rows)
    const int grp   = wid >> 1;                    // gate group 0..3 (i,f,g,o)
    const int n0g   = grp * 512 + hs;              // global gate-col base
    const int lrow0 = (wid & 1) * 16 + 8 * (lane >> 4);
    const int lcol0 = grp * 64;
    const int arow  = (wid & 1) * 16 + lm;         // row in staged A tile

    float c_reg[8];
#pragma unroll
    for (int i = 0; i < 8; ++i) c_reg[i] = 0.0f;

    for (int t = 0; t < Lq; ++t) {
        const float* hprev = (t & 1) ? hbuf1 : hbuf0;

        v8f acc[4];
#pragma unroll
        for (int nt = 0; nt < 4; ++nt) acc[nt] = (v8f){};

        // ---- stage chunk 0 (wave 0 only; all descriptor fields wave-uniform)
        // chunk ch < NCH_IH: A tile from x_t (stride L*K), B from Wih
        // chunk ch >= NCH_IH: A tile from h_{t-1} (stride H), B from Whh
        if (wid == 0) {
            for (int g = 0; g < 4; ++g)
                tdm_load_2d((unsigned)(uintptr_t)&wbuf[0][g][0],
                            wih + (size_t)0 * NGq + g * 512 + hs,
                            64, KCH, (long long)NGq, 5, 7);
            tdm_load_2d((unsigned)(uintptr_t)&abuf[0][0],
                        x + (size_t)m0wg * Lq * Kq + (size_t)t * Kq,
                        KCH, 32, (long long)Lq * Kq, 4, 3);
        }

        for (int ch = 0; ch < NCH; ++ch) {
            const int bi = ch & 1;
            if (wid == 0) {
                __builtin_amdgcn_s_wait_tensorcnt(0);  // chunk ch landed in LDS
            }
            __syncthreads();                            // publish to all waves

            if (wid == 0 && (ch + 1) < NCH) {
                const int nc = ch + 1;
                const int nb = nc & 1;
                if (nc < NCH_IH) {
                    for (int g = 0; g < 4; ++g)
                        tdm_load_2d((unsigned)(uintptr_t)&wbuf[nb][g][0],
                                    wih + (size_t)(nc * KCH) * NGq + g * 512 + hs,
                                    64, KCH, (long long)NGq, 5, 7);
                    tdm_load_2d((unsigned)(uintptr_t)&abuf[nb][0],
                                x + (size_t)m0wg * Lq * Kq + (size_t)t * Kq
                                  + nc * KCH,
                                KCH, 32, (long long)Lq * Kq, 4, 3);
                } else {
                    const int kh0 = (nc - NCH_IH) * KCH;
                    for (int g = 0; g < 4; ++g)
                        tdm_load_2d((unsigned)(uintptr_t)&wbuf[nb][g][0],
                                    whh + (size_t)kh0 * NGq + g * 512 + hs,
                                    64, KCH, (long long)NGq, 5, 7);
                    tdm_load_2d((unsigned)(uintptr_t)&abuf[nb][0],
                                hprev + (size_t)m0wg * Hq + kh0,
                                KCH, 32, (long long)Hq, 4, 3);
                }
            }

            // ---- WMMA on chunk ch, operands from LDS ----
#pragma unroll
            for (int kk = 0; kk < KCH; kk += 4) {
                v2f a = *(const v2f*)&abuf[bi][arow * AROW + kk + kh];
                const int r0 = (kk + kh) * WROW + ln;
#pragma unroll
                for (int nt = 0; nt < 4; ++nt) {
                    v2f b = { wbuf[bi][grp][r0 + nt * 16],
                              wbuf[bi][grp][r0 + WROW + nt * 16] };
                    acc[nt] = __builtin_amdgcn_wmma_f32_16x16x4_f32(
                        false, a, false, b, (short)0, acc[nt], false, false);
                }
            }
        }

        // ---- gates (+bias) -> LDS ----
#pragma unroll
        for (int nt = 0; nt < 4; ++nt) {
            float bv = bias[n0g + nt * 16 + ln];
#pragma unroll
            for (int r = 0; r < 8; ++r)
                gate_s[lrow0 + r][lcol0 + nt * 16 + ln] = acc[nt][r] + bv;
        }
        __syncthreads();

        // ---- cell update: c in registers, h -> out + double buffer ----
        float* hnext = (t & 1) ? hbuf0 : hbuf1;
#pragma unroll
        for (int i = 0; i < 8; ++i) {
            int idx = threadIdx.x + i * TPB;   // 0..2047
            int row = idx >> 6;                // 0..31
            int cc  = idx & 63;                // 0..63
            float ii = gate_s[row][cc];
            float ff = gate_s[row][64 + cc];
            float gg = gate_s[row][128 + cc];
            float oo = gate_s[row][192 + cc];
            float ig = 1.0f / (1.0f + __expf(-ii));
            float fg = 1.0f / (1.0f + __expf(-ff));
            float og = 1.0f / (1.0f + __expf(-oo));
            float gt = tanhf(gg);
            float c  = fg * c_reg[i] + ig * gt;
            c_reg[i] = c;
            float h  = og * tanhf(c);
            int b  = m0wg + row;
            int hi = hs + cc;
            out_hidden[((size_t)b * Lq + t) * Hq + hi] = h;
            hnext[b * Hq + hi] = h;
            if (t == Lq - 1) {
                out_hT[b * Hq + hi] = h;
                out_cT[b * Hq + hi] = c;
            }
        }
        __syncthreads();            // LDS reuse next step
        grid_barrier(bar, bar + 1); // publish h to all WGs
    }
}

// ---------------------------------------------------------------------------
extern "C" void kernel_launch(void* const* d_in, const int* in_sizes, int n_in,
                              void* d_out, int out_size, void* d_ws, size_t ws_size,
                              hipStream_t stream) {
    const float* x       = (const float*)d_in[0];
    const float* whh     = (const float*)d_in[1];
    const float* wih     = (const float*)d_in[2];
    const float* bias    = (const float*)d_in[3];
    const float* whh_mu  = (const float*)d_in[4];
    const float* whh_rho = (const float*)d_in[5];
    const float* wih_mu  = (const float*)d_in[6];
    const float* wih_rho = (const float*)d_in[7];
    const float* b_mu    = (const float*)d_in[8];
    const float* b_rho   = (const float*)d_in[9];

    float* out        = (float*)d_out;
    float* out_hidden = out;                                   // [B,L,H]
    float* out_loss   = out + (size_t)Bq * Lq * Hq;            // scalar
    float* out_hT     = out_loss + 1;                          // [B,H]
    float* out_cT     = out_hT + (size_t)Bq * Hq;              // [B,H]

    float* hbuf0  = (float*)d_ws;                              // [B,H]
    float* hbuf1  = hbuf0 + (size_t)Bq * Hq;                   // [B,H]
    unsigned* bar = (unsigned*)(hbuf1 + (size_t)Bq * Hq);      // 2 x u32

    init_ws_kernel<<<64, TPB, 0, stream>>>(hbuf0, bar, out_loss);
    bayes_loss_kernel<<<256, TPB, 0, stream>>>(whh, whh_mu, whh_rho,
                                               wih, wih_mu, wih_rho,
                                               bias, b_mu, b_rho, out_loss);
    blstm_kernel<<<NBLK, TPB, 0, stream>>>(x, whh, wih, bias, out_hidden,
                                           out_hT, out_cT, hbuf0, hbuf1, bar);
}